// GCN_86801289052431
// MI455X (gfx1250) — compile-verified
//
#include <hip/hip_runtime.h>

#define D 64

typedef __attribute__((ext_vector_type(2))) float v2f;
typedef __attribute__((ext_vector_type(8))) float v8f;

// ---------------- degree kernels ----------------

__global__ void deg_init(float* __restrict__ deg, int n) {
    int i = blockIdx.x * blockDim.x + threadIdx.x;
    if (i < n) deg[i] = 1.0f;              // self-loop contributes 1
}

__global__ void deg_count(const int* __restrict__ dst, float* __restrict__ deg, int e) {
    int i = blockIdx.x * blockDim.x + threadIdx.x;
    if (i < e) atomicAdd(&deg[dst[i]], 1.0f);
}

__global__ void deg_finalize(float* __restrict__ deg, int n) {
    int i = blockIdx.x * blockDim.x + threadIdx.x;
    if (i < n) deg[i] = rsqrtf(deg[i]);    // deg >= 1 always (self-loops)
}

// ---------------- WMMA fp32 GEMM + fused self-loop/bias epilogue ----------------
//   h   = act(in) @ W                       -> h_out  (raw, needed by edge gather)
//   agg = dis[row]^2 * h + bias             -> agg_out (self-loop + bias init)
// in: [n,64], W: [64,64] row-major. Block = 256 threads = 8 waves; block covers
// 32 rows; each wave owns one 16x16 fp32 C tile (wave>>2 row tile, wave&3 col tile).
template <bool RELU>
__global__ __launch_bounds__(256) void gemm64_wmma(const float* __restrict__ in,
                                                   const float* __restrict__ W,
                                                   const float* __restrict__ dis,
                                                   const float* __restrict__ bias,
                                                   float* __restrict__ h_out,
                                                   float* __restrict__ agg_out,
                                                   int n) {
    __shared__ float sW[D * D];            // 16 KB of 320 KB LDS
    for (int i = threadIdx.x; i < D * D; i += 256) sW[i] = W[i];
    __syncthreads();

    const int wave  = threadIdx.x >> 5;
    const int lane  = threadIdx.x & 31;
    const int lhalf = lane & 15;
    const int khi   = (lane >> 4) * 2;     // 0 for lanes 0-15, 2 for lanes 16-31

    const int r0   = (blockIdx.x * 2 + (wave >> 2)) * 16;  // row tile base
    const int col0 = (wave & 3) * 16;                       // col tile base
    if (r0 >= n) return;                   // wave-uniform exit (EXEC stays full)

    int arow = r0 + lhalf;
    if (arow >= n) arow = n - 1;           // clamp, keep all lanes active
    const float* __restrict__ aptr = in + (size_t)arow * D + khi;

    v8f acc = {};
#pragma unroll
    for (int k = 0; k < D; k += 4) {
        v2f a;
        a.x = aptr[k];
        a.y = aptr[k + 1];
        if (RELU) { a.x = fmaxf(a.x, 0.0f); a.y = fmaxf(a.y, 0.0f); }
        v2f b;
        b.x = sW[(k + khi    ) * D + col0 + lhalf];
        b.y = sW[(k + khi + 1) * D + col0 + lhalf];
        // D = A(16x4) * B(4x16) + C ; fp32 in / fp32 acc
        acc = __builtin_amdgcn_wmma_f32_16x16x4_f32(
            /*neg_a=*/false, a, /*neg_b=*/false, b,
            /*c_mod=*/(short)0, acc, /*reuse_a=*/false, /*reuse_b=*/false);
    }

    // C layout: VGPR r -> row r (lanes 0-15) / row r+8 (lanes 16-31), col = lane%16
    const int   rbase = r0 + (lane >> 4) * 8;
    const int   c     = col0 + lhalf;
    const float bc    = bias[c];
#pragma unroll
    for (int r = 0; r < 8; ++r) {
        int row = rbase + r;
        if (row < n) {
            float hv = acc[r];
            float dv = dis[row];                       // lane-broadcast load
            size_t idx = (size_t)row * D + c;
            h_out[idx]   = hv;
            agg_out[idx] = hv * (dv * dv) + bc;        // self-loop + bias
        }
    }
}

// ---------------- edge scatter-add:  out[dst] += dis[src]*dis[dst] * h[src] ----------------
__global__ void edge_aggregate(const float* __restrict__ h, const int* __restrict__ src,
                               const int* __restrict__ dst, const float* __restrict__ dis,
                               float* __restrict__ out, int e) {
    int t  = blockIdx.x * blockDim.x + threadIdx.x;
    int ed = t >> 6;
    int ch = t & 63;
    if (ed >= e) return;
    int s = src[ed];
    int d = dst[ed];
    float w = dis[s] * dis[d];
    float v = h[(size_t)s * D + ch] * w;
    atomicAdd(&out[(size_t)d * D + ch], v);
}

// ---------------- launch ----------------

extern "C" void kernel_launch(void* const* d_in, const int* in_sizes, int n_in,
                              void* d_out, int out_size, void* d_ws, size_t ws_size,
                              hipStream_t stream) {
    const float* x  = (const float*)d_in[0];
    const int*   ei = (const int*)d_in[1];
    const float* W1 = (const float*)d_in[2];
    const float* b1 = (const float*)d_in[3];
    const float* W2 = (const float*)d_in[4];
    const float* b2 = (const float*)d_in[5];
    const float* W3 = (const float*)d_in[6];
    const float* b3 = (const float*)d_in[7];

    const int n = in_sizes[0] / D;
    const int e = in_sizes[1] / 2;
    const int* src = ei;          // edge_index[0]
    const int* dst = ei + e;      // edge_index[1]

    float* H   = (float*)d_ws;                  // [n,64] gemm output (raw h)
    float* AGG = H   + (size_t)n * D;           // [n,64] aggregated layer output
    float* DIS = AGG + (size_t)n * D;           // [n]    deg^{-1/2}
    float* OUT = (float*)d_out;

    const int TB  = 256;
    const int gN  = (n + TB - 1) / TB;
    const int gE  = (e + TB - 1) / TB;
    const int gEC = (int)(((size_t)e * D + TB - 1) / TB);
    const int gG  = (n + 31) / 32;              // 32 rows per gemm block

    // degrees -> deg^{-1/2}
    deg_init<<<gN, TB, 0, stream>>>(DIS, n);
    deg_count<<<gE, TB, 0, stream>>>(dst, DIS, e);
    deg_finalize<<<gN, TB, 0, stream>>>(DIS, n);

    // layer 1: h = x@W1 ; agg initialized to self-loop+bias in gemm epilogue
    gemm64_wmma<false><<<gG, TB, 0, stream>>>(x, W1, DIS, b1, H, AGG, n);
    edge_aggregate<<<gEC, TB, 0, stream>>>(H, src, dst, DIS, AGG, e);

    // layer 2: relu of previous agg fused into gemm A-read
    gemm64_wmma<true><<<gG, TB, 0, stream>>>(AGG, W2, DIS, b2, H, AGG, n);
    edge_aggregate<<<gEC, TB, 0, stream>>>(H, src, dst, DIS, AGG, e);

    // layer 3: final output, no trailing relu; epilogue writes d_out directly
    gemm64_wmma<true><<<gG, TB, 0, stream>>>(AGG, W3, DIS, b3, H, OUT, n);
    edge_aggregate<<<gEC, TB, 0, stream>>>(H, src, dst, DIS, OUT, e);
}